// LNon_16621523436082
// MI455X (gfx1250) — compile-verified
//
#include <hip/hip_runtime.h>
#include <math.h>

#ifndef __has_builtin
#define __has_builtin(x) 0
#endif

#define AS1 __attribute__((address_space(1)))
#define AS3 __attribute__((address_space(3)))

// Async global->LDS staging path (CDNA5). Guarded so it compiles on host pass
// and on toolchains lacking the builtin; falls back to direct B128 loads.
#if defined(__HIP_DEVICE_COMPILE__) && __has_builtin(__builtin_amdgcn_global_load_async_to_lds_b128)
#define USE_ASYNC_LDS 1
#else
#define USE_ASYNC_LDS 0
#endif

__device__ __forceinline__ void wait_async_le1() {
#if defined(__HIP_DEVICE_COMPILE__)
#if __has_builtin(__builtin_amdgcn_s_wait_asynccnt)
    __builtin_amdgcn_s_wait_asynccnt(1);
#else
    asm volatile("s_wait_asynccnt 1" ::: "memory");
#endif
#endif
}
__device__ __forceinline__ void wait_async_le0() {
#if defined(__HIP_DEVICE_COMPILE__)
#if __has_builtin(__builtin_amdgcn_s_wait_asynccnt)
    __builtin_amdgcn_s_wait_asynccnt(0);
#else
    asm volatile("s_wait_asynccnt 0" ::: "memory");
#endif
#endif
}

#if USE_ASYNC_LDS
// Builtin signature (from compiler diagnostic): (AS1 v4i*, AS3 v4i*, imm, imm)
typedef int v4i __attribute__((vector_size(4 * sizeof(int))));

__device__ __forceinline__ void async_ld16(const float4* g, float4* lds) {
    __builtin_amdgcn_global_load_async_to_lds_b128(
        (AS1 v4i*)(unsigned long long)(const void*)g,  // global ptr: bitwise-identical
        (AS3 v4i*)(void*)lds,                          // generic->LDS addrspacecast
        0, 0);
}
#endif

// ---- monotone ordered-uint encoding for float atomic min/max ----
__device__ __forceinline__ unsigned f2key(float f) {
    unsigned u = __float_as_uint(f);
    return (u & 0x80000000u) ? ~u : (u | 0x80000000u);
}
__device__ __forceinline__ float key2f(unsigned k) {
    unsigned u = (k & 0x80000000u) ? (k ^ 0x80000000u) : ~k;
    return __uint_as_float(u);
}

// ws layout:  u32[0]=minkey(raw)  u32[1]=maxkey(raw)  u32[2..6]=counts[5]
// float table at byte offset 64:
//   T[0]=grid0 T[1]=width T[2..5]=segSlope T[6..9]=segIntercept
//   T[10..14]=frame+0.001*theta  T[15..19]=frame+0.001*velocity

__global__ void k_init(unsigned* mm) {
    int t = threadIdx.x;
    if (t == 0) mm[0] = 0xFFFFFFFFu;
    else if (t == 1) mm[1] = 0u;
    else if (t < 7) mm[t] = 0u;
}

// Min/max of RAW data (scale by channel_transform applied later, sign-aware).
__global__ void __launch_bounds__(256) k_minmax(const float* __restrict__ data,
                                                unsigned* __restrict__ mm, int n) {
    __shared__ float smn[256], smx[256];
    float lmin = 3.4e38f, lmax = -3.4e38f;
    const int stride = gridDim.x * blockDim.x;
    const int n4 = n >> 2;
    const float4* d4 = (const float4*)data;
    for (int i = blockIdx.x * blockDim.x + threadIdx.x; i < n4; i += stride) {
        if (i + stride < n4) __builtin_prefetch(&d4[i + stride], 0, 0);
        float4 v = d4[i];
        lmin = fminf(lmin, fminf(fminf(v.x, v.y), fminf(v.z, v.w)));
        lmax = fmaxf(lmax, fmaxf(fmaxf(v.x, v.y), fmaxf(v.z, v.w)));
    }
    for (int i = (n4 << 2) + blockIdx.x * blockDim.x + threadIdx.x; i < n; i += stride) {
        float a = data[i];
        lmin = fminf(lmin, a);
        lmax = fmaxf(lmax, a);
    }
    smn[threadIdx.x] = lmin;
    smx[threadIdx.x] = lmax;
    __syncthreads();
    for (int off = 128; off > 0; off >>= 1) {
        if (threadIdx.x < off) {
            smn[threadIdx.x] = fminf(smn[threadIdx.x], smn[threadIdx.x + off]);
            smx[threadIdx.x] = fmaxf(smx[threadIdx.x], smx[threadIdx.x + off]);
        }
        __syncthreads();
    }
    if (threadIdx.x == 0) {
        atomicMin(&mm[0], f2key(smn[0]));
        atomicMax(&mm[1], f2key(smx[0]));
    }
}

// scaled min/max from raw min/max + channel scale (sign-aware swap)
__device__ __forceinline__ void scaled_range(float rmin, float rmax, float cs,
                                             float& smin, float& smax) {
    smin = cs >= 0.f ? cs * rmin : cs * rmax;
    smax = cs >= 0.f ? cs * rmax : cs * rmin;
}

__global__ void __launch_bounds__(256) k_hist(const float* __restrict__ data,
                                              const float* __restrict__ cts,
                                              const unsigned* __restrict__ mm,
                                              unsigned* __restrict__ counts, int n) {
    __shared__ unsigned sc[5];
    if (threadIdx.x < 5) sc[threadIdx.x] = 0u;
    __syncthreads();
    const float cs = cts[0];
    float smin, smax;
    scaled_range(key2f(mm[0]), key2f(mm[1]), cs, smin, smax);
    const float dmin = smin - 0.1f;
    const float dmax = smax + 0.1f;
    const float invw = 5.0f / (dmax - dmin);
    // bin(x) = floor((x*cs - dmin) * invw) == floor(fma(x, cs*invw, -dmin*invw))
    const float scale = cs * invw;
    const float bias = -dmin * invw;

    unsigned c0 = 0, c1 = 0, c2 = 0, c3 = 0, c4 = 0;   // per-lane bin counters
    unsigned pck = 0;                                   // 5 x 6-bit packed fields
    int iter = 0;
    const int stride = gridDim.x * blockDim.x;
    const int n4 = n >> 2;
    const float4* d4 = (const float4*)data;

    for (int i = blockIdx.x * blockDim.x + threadIdx.x; i < n4; i += stride) {
        float4 v = d4[i];
        float e[4] = {v.x, v.y, v.z, v.w};
#pragma unroll
        for (int c = 0; c < 4; ++c) {
            int b = (int)floorf(fmaf(e[c], scale, bias));
            b = b < 0 ? 0 : (b > 4 ? 4 : b);
            pck += 1u << (b * 6);                       // no LDS-atomic serialization
        }
        if (++iter == 8) {                              // 32 elems < 63 per field: flush
            c0 += pck & 63u; c1 += (pck >> 6) & 63u; c2 += (pck >> 12) & 63u;
            c3 += (pck >> 18) & 63u; c4 += (pck >> 24) & 63u;
            pck = 0; iter = 0;
        }
    }
    for (int i = (n4 << 2) + blockIdx.x * blockDim.x + threadIdx.x; i < n; i += stride) {
        int b = (int)floorf(fmaf(data[i], scale, bias));
        b = b < 0 ? 0 : (b > 4 ? 4 : b);
        pck += 1u << (b * 6);
    }
    c0 += pck & 63u; c1 += (pck >> 6) & 63u; c2 += (pck >> 12) & 63u;
    c3 += (pck >> 18) & 63u; c4 += (pck >> 24) & 63u;

    if (c0) atomicAdd(&sc[0], c0);
    if (c1) atomicAdd(&sc[1], c1);
    if (c2) atomicAdd(&sc[2], c2);
    if (c3) atomicAdd(&sc[3], c3);
    if (c4) atomicAdd(&sc[4], c4);
    __syncthreads();
    if (threadIdx.x < 5) atomicAdd(&counts[threadIdx.x], sc[threadIdx.x]);
}

__global__ void k_prep(const float* __restrict__ params, const float* __restrict__ cts,
                       const unsigned* __restrict__ mm,
                       const unsigned* __restrict__ counts, float* __restrict__ T, int n) {
    if (blockIdx.x != 0 || threadIdx.x != 0) return;
    const float cs = cts[0];
    float smin, smax;
    scaled_range(key2f(mm[0]), key2f(mm[1]), cs, smin, smax);
    float dmin = smin - 0.1f;
    float dmax = smax + 0.1f;
    float width = (dmax - dmin) * 0.2f;
    float accum[5];
    float cum = 0.f;
    float invN = 1.0f / (float)n;
    for (int i = 0; i < 5; ++i) { cum += (float)counts[i] * invN; accum[i] = cum * 5.0f; }
    float grid[5];
    for (int i = 0; i < 5; ++i) grid[i] = dmin + width * ((float)i + 0.5f);
    float frame[5];
    for (int q = 0; q < 5; ++q) {
        float qf = (float)q;
        int ss = 0;
        while (ss < 5 && accum[ss] < qf) ++ss;   // searchsorted(accum, qf, 'left')
        int idx = ss - 1;
        idx = idx < 0 ? 0 : (idx > 3 ? 3 : idx);
        float den = accum[idx + 1] - accum[idx];
        if (den == 0.f) den = 1.f;
        frame[q] = grid[idx] + (grid[idx + 1] - grid[idx]) / den * (qf - accum[idx]);
    }
    T[0] = grid[0];
    T[1] = width;
    for (int i = 0; i < 4; ++i) {
        float S = (accum[i + 1] - accum[i]) / width;  // grid is uniform: denom == width
        T[2 + i] = S;
        T[6 + i] = accum[i] - S * grid[i];
    }
    for (int i = 0; i < 5; ++i) {
        T[10 + i] = frame[i] + 0.001f * params[i];      // theta row
        T[15 + i] = frame[i] + 0.001f * params[5 + i];  // velocity row
    }
}

struct Tab {
    float cs, ss, invw, nb;   // nb = -grid0*invw, so seg = floor(fma(d, invw, nb))
};

__device__ __forceinline__ float proc(float x, const Tab& t,
                                      const float2* s_seg, const float2* s_tv) {
    float d = x * t.cs;
    int i = (int)floorf(fmaf(d, t.invw, t.nb));
    i = i < 0 ? 0 : (i > 3 ? 3 : i);
    float2 sg = s_seg[i];                         // ds_load_b64 gather {slope, intercept}
    float index = fmaf(sg.x, d, sg.y);
    float bf = floorf(index);
    float pos = index - bf;
    int bi = (int)bf;
    int b = bi < 0 ? 0 : (bi > 4 ? 4 : bi);
    int e = b + 1 > 4 ? 4 : b + 1;
    float2 pb = s_tv[b], pe = s_tv[e];            // ds_load_b64 gathers {theta, velo}
    float th = fmaf(pos, pe.x - pb.x, pb.x);      // (1-pos)*p[b] + pos*p[e]
    float vl = fmaf(pos, pe.y - pb.y, pb.y);
    float sn, cn;
    __sincosf(th, &sn, &cn);                      // v_sin_f32 / v_cos_f32 (TRANS)
    float val = fmaf(d, fmaf(vl, sn, 1.0f), vl * cn);  // d*(1+v*sin)+v*cos
    return val * t.ss;
}

__global__ void __launch_bounds__(256) k_main(const float* __restrict__ data,
                                              const float* __restrict__ cts,
                                              const float* __restrict__ sts,
                                              const float* __restrict__ T,
                                              float* __restrict__ out, int n) {
    __shared__ float2 s_tv[5];    // {theta_tab, velo_tab}
    __shared__ float2 s_seg[4];   // {slope, intercept}
#if USE_ASYNC_LDS
    __shared__ float4 stage[2][256];
#endif
    if (threadIdx.x < 5) {
        float2 tv; tv.x = T[10 + threadIdx.x]; tv.y = T[15 + threadIdx.x];
        s_tv[threadIdx.x] = tv;
    } else if (threadIdx.x >= 8 && threadIdx.x < 12) {
        int k = threadIdx.x - 8;
        float2 sg; sg.x = T[2 + k]; sg.y = T[6 + k];
        s_seg[k] = sg;
    }
    Tab t;
    t.cs = cts[0];
    t.ss = sts[0];
    float grid0 = T[0];
    t.invw = 1.0f / T[1];
    t.nb = -grid0 * t.invw;
    __syncthreads();

    const int n4 = n >> 2;
    const float4* d4 = (const float4*)data;
    float4* o4 = (float4*)out;

#if USE_ASYNC_LDS
    const int nTiles = (n4 + 255) >> 8;
    int tt = blockIdx.x;
    {
        int i4 = tt * 256 + threadIdx.x;
        if (tt < nTiles && i4 < n4) async_ld16(&d4[i4], &stage[0][threadIdx.x]);
    }
    int buf = 0;
    for (; tt < nTiles; tt += gridDim.x) {
        int ttn = tt + gridDim.x;
        bool more = (ttn < nTiles);
        if (more) {
            int i4n = ttn * 256 + threadIdx.x;
            if (i4n < n4) async_ld16(&d4[i4n], &stage[buf ^ 1][threadIdx.x]);
            wait_async_le1();   // in-order async completion: current tile is in LDS
        } else {
            wait_async_le0();
        }
        asm volatile("" ::: "memory");
        int i4 = tt * 256 + threadIdx.x;
        if (i4 < n4) {
            float4 v = stage[buf][threadIdx.x];  // ds_load_b128, own lane's slot
            float4 r;
            r.x = proc(v.x, t, s_seg, s_tv);
            r.y = proc(v.y, t, s_seg, s_tv);
            r.z = proc(v.z, t, s_seg, s_tv);
            r.w = proc(v.w, t, s_seg, s_tv);
            o4[i4] = r;
        }
        buf ^= 1;
    }
#else
    const int stride = gridDim.x * blockDim.x;
    for (int i4 = blockIdx.x * blockDim.x + threadIdx.x; i4 < n4; i4 += stride) {
        float4 v = d4[i4];
        float4 r;
        r.x = proc(v.x, t, s_seg, s_tv);
        r.y = proc(v.y, t, s_seg, s_tv);
        r.z = proc(v.z, t, s_seg, s_tv);
        r.w = proc(v.w, t, s_seg, s_tv);
        o4[i4] = r;
    }
#endif
    // scalar tail (n % 4 elements)
    if (blockIdx.x == 0 && threadIdx.x == 0) {
        for (int i = n4 << 2; i < n; ++i) out[i] = proc(data[i], t, s_seg, s_tv);
    }
}

extern "C" void kernel_launch(void* const* d_in, const int* in_sizes, int n_in,
                              void* d_out, int out_size, void* d_ws, size_t ws_size,
                              hipStream_t stream) {
    const float* data = (const float*)d_in[0];
    const float* params = (const float*)d_in[1];   // 10 floats: theta[5], velocity[5]
    const float* cts = (const float*)d_in[2];      // scalar
    const float* sts = (const float*)d_in[3];      // scalar
    float* out = (float*)d_out;
    int n = in_sizes[0];

    unsigned* mm = (unsigned*)d_ws;                   // [0]=minkey [1]=maxkey [2..6]=counts
    float* T = (float*)((char*)d_ws + 64);            // 20-float table

    const int blocks = 4096;
    k_init<<<1, 32, 0, stream>>>(mm);
    k_minmax<<<blocks, 256, 0, stream>>>(data, mm, n);
    k_hist<<<blocks, 256, 0, stream>>>(data, cts, mm, mm + 2, n);
    k_prep<<<1, 1, 0, stream>>>(params, cts, mm, mm + 2, T, n);
    k_main<<<blocks, 256, 0, stream>>>(data, cts, sts, T, out, n);
}